// SpikeInnerProduct_7181185319483
// MI455X (gfx1250) — compile-verified
//
#include <hip/hip_runtime.h>

typedef __attribute__((ext_vector_type(2))) float v2f;
typedef __attribute__((ext_vector_type(4))) float v4f;
typedef __attribute__((ext_vector_type(8))) float v8f;

#define T_STEPS 63
#define HEADS   16
#define SDIM    384
#define DDIM    128
#define SLAB    (HEADS * SDIM * DDIM)   /* 786432 floats per timestep, same for x and y */
#define SLAB4   (SLAB / 4)              /* 196608 float4 */
#define WSLAB   ((size_t)HEADS * SDIM * SDIM) /* 2359296 floats: one w image */

// ---------------------------------------------------------------------------
// Kernel 1: temporal mean over 63 steps for BOTH x and y in one pass.
//   xb -> ws[0 .. SLAB)        laid out [H][S][D]  (row-major, K contiguous)
//   yb -> ws[SLAB .. 2*SLAB)   laid out [H][D][S]  (row-major, N contiguous)
// x/y are read exactly once -> non-temporal loads (TH=NT) so the 396 MB
// stream does not evict the 6 MB xb/yb panels that kernel 2 reuses from L2.
// ---------------------------------------------------------------------------
__global__ void temporal_mean_kernel(const float* __restrict__ x,
                                     const float* __restrict__ y,
                                     float* __restrict__ ws) {
    int j = blockIdx.x * blockDim.x + threadIdx.x;   // 0 .. 2*SLAB4-1
    const v4f* __restrict__ src;
    v4f* __restrict__ dst;
    int off;
    if (j < SLAB4) {
        src = (const v4f*)x;
        dst = (v4f*)ws;
        off = j;
    } else {
        src = (const v4f*)y;
        dst = (v4f*)ws + SLAB4;
        off = j - SLAB4;
    }
    v4f acc = {0.f, 0.f, 0.f, 0.f};
    #pragma unroll 7
    for (int t = 0; t < T_STEPS; ++t) {
        acc += __builtin_nontemporal_load(&src[(size_t)t * SLAB4 + off]);
    }
    const float inv = 1.0f / 63.0f;
    acc *= inv;
    dst[off] = acc;   // regular store: ws stays L2-resident for kernel 2
}

// ---------------------------------------------------------------------------
// Kernel 2: per-head fp32 WMMA GEMM w[h] = xb[h] @ yb[h], fused with the
// 63-way temporal broadcast (tile stays in VGPRs; stored 63 times -> w never
// touches HBM as an intermediate).
//
// One wave (32 lanes) per 16x16 output tile. K=128 -> 32x v_wmma_f32_16x16x4.
// ISA layouts (CDNA5 ISA):
//   A 16x4 f32 : lanes 0-15 hold M=lane, K={0,1}; lanes 16-31 hold K={2,3}
//   B 4x16 f32 : lanes 0-15 hold N=lane, K={0,1}; lanes 16-31 hold K={2,3}
//   C/D 16x16  : VGPR r, lanes 0-15 -> (M=r,   N=lane)
//                        lanes 16-31 -> (M=r+8, N=lane-16)
// Output is a write-once 594 MB stream -> non-temporal stores.
// ---------------------------------------------------------------------------
__global__ void __launch_bounds__(32)
gemm_bcast_kernel(const float* __restrict__ ws, float* __restrict__ out) {
    const int lane = threadIdx.x;        // 0..31, full wave, EXEC all ones
    const int half = lane >> 4;          // 0 or 1
    const int lm   = lane & 15;

    const int h  = blockIdx.z;           // head 0..15
    const int mb = blockIdx.y << 4;      // tile row base in S (0..368)
    const int nb = blockIdx.x << 4;      // tile col base in S (0..368)

    const float* __restrict__ xb = ws;                 // [H][S][D]
    const float* __restrict__ yb = ws + SLAB;          // [H][D][S]

    // Per-lane base pointers
    const float* Arow = xb + ((size_t)h * SDIM + (mb + lm)) * DDIM; // A row (K contiguous)
    const float* Bcol = yb + (size_t)h * DDIM * SDIM + (nb + lm);   // B column (stride SDIM)

    v8f c = {0.f, 0.f, 0.f, 0.f, 0.f, 0.f, 0.f, 0.f};

    #pragma unroll
    for (int k = 0; k < DDIM; k += 4) {
        const int ka = k + 2 * half;     // this lane's K pair
        // A fragment: two consecutive K values -> one b64 load
        v2f a = *(const v2f*)(Arow + ka);
        // B fragment: two K values of one column, stride SDIM
        v2f b;
        b.x = Bcol[(size_t)ka * SDIM];
        b.y = Bcol[(size_t)(ka + 1) * SDIM];
        c = __builtin_amdgcn_wmma_f32_16x16x4_f32(
                /*neg_a=*/false, a, /*neg_b=*/false, b,
                /*c_mod=*/(short)0, c, /*reuse_a=*/false, /*reuse_b=*/false);
    }

    // Fused broadcast epilogue: write this 16x16 tile into all 63 time slices.
    // out index: t*WSLAB + h*S*S + m*S + n. Streaming write-once -> NT stores.
    float* obase = out + ((size_t)h * SDIM + mb) * SDIM + nb;
    for (int t = 0; t < T_STEPS; ++t) {
        float* o = obase + (size_t)t * WSLAB;
        #pragma unroll
        for (int r = 0; r < 8; ++r) {
            __builtin_nontemporal_store(c[r],
                &o[(size_t)(r + 8 * half) * SDIM + lm]);
        }
    }
}

extern "C" void kernel_launch(void* const* d_in, const int* in_sizes, int n_in,
                              void* d_out, int out_size, void* d_ws, size_t ws_size,
                              hipStream_t stream) {
    const float* x = (const float*)d_in[0];   // [63,1,16,384,128] fp32
    const float* y = (const float*)d_in[1];   // [63,1,16,128,384] fp32
    float* out = (float*)d_out;               // [63,1,16,384,384] fp32
    float* ws  = (float*)d_ws;                // needs 2*SLAB*4 = 6 MB

    // Kernel 1: 2*SLAB4 = 393216 float4 outputs, 256 threads/block
    temporal_mean_kernel<<<(2 * SLAB4) / 256, 256, 0, stream>>>(x, y, ws);

    // Kernel 2: 24x24 tiles per head, 16 heads, one wave per tile
    dim3 grid(SDIM / 16, SDIM / 16, HEADS);
    gemm_bcast_kernel<<<grid, 32, 0, stream>>>(ws, out);
}